// LocallyConnected2D_44925357916366
// MI455X (gfx1250) — compile-verified
//
#include <hip/hip_runtime.h>
#include <stdint.h>

typedef __attribute__((ext_vector_type(2))) float v2f;
typedef __attribute__((ext_vector_type(8))) float v8f;

#define B_   32
#define CIN  64
#define COUT 64
#define IH   32
#define IW   32
#define OH   30
#define OW   30
#define NPIX (OH * OW)          // 900
#define KTOT (CIN * 9)          // 576 = C_IN*KH*KW
#define KC   64                 // K chunk staged in LDS
#define LDA  66                 // padded row stride for As (conflict-free, b64-alignable)
#define LDB  64

__global__ __launch_bounds__(256)
void LocallyConnected2D_kernel(const float* __restrict__ x,
                               const float* __restrict__ wt,
                               const float* __restrict__ bias,
                               float* __restrict__ out) {
  // double-buffered staging: 2 * (8.4KB + 16KB) ~ 50KB of the 320KB WGP LDS
  __shared__ float As[2][B_ * LDA];   // [32][66] patches  (A: M=batch, K)
  __shared__ float Bs[2][KC * LDB];   // [64][64] weights  (B: K, N=c_out)

  const int pix  = blockIdx.x;       // 0..899
  const int h    = pix / OW;
  const int wpix = pix - h * OW;
  const int tid  = threadIdx.x;      // 0..255
  const int lane = tid & 31;
  const int wave = tid >> 5;         // 0..7
  const int mt   = wave >> 2;        // batch tile 0..1
  const int nt   = wave & 3;         // c_out tile 0..3
  const int lhalf = lane >> 4;       // 0: lanes 0-15, 1: lanes 16-31
  const int l15   = lane & 15;

  // per-thread gather assignments
  const int ab   = tid & 31;         // batch row this thread stages (A)
  const int akl0 = (tid >> 5) * 8;   // 8 A elements / thread (2048 total)
  const int bo   = tid & 63;         // c_out column this thread stages (B)
  const int bkl0 = (tid >> 6) * 16;  // 16 B elements / thread (4096 total)

  const float* xb = x + ((ab * CIN) * IH + h) * IW + wpix; // + (c*IH + p)*IW + q
  const float* wb = wt + (size_t)bo * KTOT * NPIX + pix;   // + k*NPIX

  // Issue the async global->LDS gather for one K chunk into buffer `buf`.
  // global_load_async_to_lds_b32: VDST = per-lane LDS byte address,
  // VADDR = per-lane 64-bit global address (GV mode). Tracked by ASYNCcnt.
  auto issue = [&](int kc, int buf) {
#pragma unroll
    for (int i = 0; i < 8; ++i) {
      const int kl = akl0 + i;
      const int k  = kc + kl;
      const int c  = k / 9;
      const int r  = k - c * 9;
      const int p  = r / 3;
      const int q  = r - p * 3;
      const float* g = xb + (c * IH + p) * IW + q;
      // low 32 bits of the generic pointer == wave-relative LDS byte offset
      const uint32_t l = (uint32_t)(uintptr_t)&As[buf][ab * LDA + kl];
      asm volatile("global_load_async_to_lds_b32 %0, %1, off"
                   :: "v"(l), "v"(g) : "memory");
    }
#pragma unroll
    for (int i = 0; i < 16; ++i) {
      const int kl = bkl0 + i;
      const int k  = kc + kl;
      const float* g = wb + (size_t)k * NPIX;
      const uint32_t l = (uint32_t)(uintptr_t)&Bs[buf][kl * LDB + bo];
      asm volatile("global_load_async_to_lds_b32 %0, %1, off"
                   :: "v"(l), "v"(g) : "memory");
    }
  };

  v8f acc = {};
  issue(0, 0);                       // prologue: chunk 0 -> buffer 0
  int buf = 0;

  for (int kc = 0; kc < KTOT; kc += KC) {
    // my chunk-kc async loads complete, then workgroup-wide rendezvous:
    // after the barrier, everyone's chunk-kc data is in LDS and everyone
    // has finished computing on the other buffer.
    asm volatile("s_wait_asynccnt 0x0" ::: "memory");
    __syncthreads();

    if (kc + KC < KTOT) issue(kc + KC, buf ^ 1);  // overlap next gather

    const float* Ab = &As[buf][0];
    const float* Bb = &Bs[buf][0];
    // 16 x V_WMMA_F32_16X16X4_F32 over this K chunk
#pragma unroll
    for (int kk = 0; kk < KC; kk += 4) {
      const int ke = kk + lhalf * 2;  // lanes 0-15: K={kk,kk+1}; 16-31: K={kk+2,kk+3}
      v2f a, bf;
      a.x  = Ab[(mt * 16 + l15) * LDA + ke];
      a.y  = Ab[(mt * 16 + l15) * LDA + ke + 1];
      bf.x = Bb[(ke + 0) * LDB + nt * 16 + l15];
      bf.y = Bb[(ke + 1) * LDB + nt * 16 + l15];
      acc = __builtin_amdgcn_wmma_f32_16x16x4_f32(
          /*neg_a=*/false, a, /*neg_b=*/false, bf,
          /*c_mod=*/(short)0, acc, /*reuse_a=*/false, /*reuse_b=*/false);
    }
    buf ^= 1;
  }

  // ---- epilogue: bias + scatter-store per C/D layout
  // VGPR r: lanes 0-15 -> M=r, lanes 16-31 -> M=r+8; N = lane&15
  const int o  = nt * 16 + l15;
  const float bv = bias[o * NPIX + pix];
#pragma unroll
  for (int r = 0; r < 8; ++r) {
    const int b = mt * 16 + r + lhalf * 8;
    out[(b * COUT + o) * NPIX + pix] = acc[r] + bv;
  }
}

extern "C" void kernel_launch(void* const* d_in, const int* in_sizes, int n_in,
                              void* d_out, int out_size, void* d_ws, size_t ws_size,
                              hipStream_t stream) {
  const float* x    = (const float*)d_in[0];  // [32,64,32,32]
  const float* wt   = (const float*)d_in[1];  // [64,64,3,3,30,30]
  const float* bias = (const float*)d_in[2];  // [64,30,30]
  float* out = (float*)d_out;                 // [32,64,30,30]

  dim3 grid(NPIX);   // one workgroup per output pixel
  dim3 block(256);   // 8 waves: 2 batch-tiles x 4 cout-tiles
  LocallyConnected2D_kernel<<<grid, block, 0, stream>>>(x, wt, bias, out);
}